// Attn_1898375545663
// MI455X (gfx1250) — compile-verified
//
#include <hip/hip_runtime.h>
#include <hip/hip_bf16.h>
#include <math.h>

// Problem dims (fixed by the reference)
#define L_DIM 4096
#define B_DIM 32
#define H_DIM 1024

typedef __attribute__((ext_vector_type(2))) float v2f;
typedef __attribute__((ext_vector_type(8))) float v8f;

// ---------------------------------------------------------------------------
// Pass 1: energies[b, l] = sum_h hidden[h, b] * enc[l, b, h]
// One wave per (b, 16-wide l tile). WMMA f32 16x16x4, accumulate over h.
//   A[m, k] = enc[l0+m, b, h0+k]          (16x4, fp32)
//   B[k, n] = hidden[h0+k, bg+n]          (4x16, fp32), bg = (b/16)*16
//   D[m, n] accumulates; only column n* = b-bg is the wanted GEMV result.
// Blocks of 256 threads = 8 waves; 1024 blocks -> 8192 waves = 32 b * 256 tiles.
// ---------------------------------------------------------------------------
__global__ __launch_bounds__(256)
void energies_wmma_kernel(const float* __restrict__ hidden,
                          const float* __restrict__ enc,
                          float* __restrict__ energies) {
    const int lane = threadIdx.x & 31;
    const int waveInBlock = threadIdx.x >> 5;
    const int w = blockIdx.x * 8 + waveInBlock;   // global wave id
    const int b = w >> 8;                          // 0..31
    const int ltile = w & 255;                     // 0..255
    const int l0 = ltile << 4;
    const int bg = b & ~15;                        // batch group base for B cols

    const int mn  = lane & 15;                     // M for A rows, N for B cols
    const int hi  = lane >> 4;                     // 0 or 1
    const int kb  = hi << 1;                       // this lane's K base: 0 or 2

    // A: lane holds enc row (l0+mn, b), elements h0+kb, h0+kb+1 each step.
    const float* __restrict__ arow =
        enc + ((size_t)(l0 + mn) * B_DIM + b) * H_DIM;
    // B: lane holds hidden column (bg+mn), elements (h0+kb)*B, (h0+kb+1)*B.
    const float* __restrict__ bcol = hidden + (bg + mn);

    v8f acc = {0.f, 0.f, 0.f, 0.f, 0.f, 0.f, 0.f, 0.f};

#pragma unroll 4
    for (int h0 = 0; h0 < H_DIM; h0 += 4) {
        // 8-byte aligned contiguous pair along h
        v2f a = *(const v2f*)(arow + h0 + kb);
        v2f bv;
        bv.x = bcol[(size_t)(h0 + kb) * B_DIM];
        bv.y = bcol[(size_t)(h0 + kb + 1) * B_DIM];
        // D = A * B + C   (8-arg form: neg_a, A, neg_b, B, c_mod, C, reuse_a, reuse_b)
        acc = __builtin_amdgcn_wmma_f32_16x16x4_f32(
            false, a, false, bv, (short)0, acc, false, false);
    }

    // Extract the valid column n* = b - bg.
    // D layout: VGPR r, lanes 0-15 -> (M=r, N=lane); lanes 16-31 -> (M=r+8, N=lane-16).
    const int nstar = b & 15;
    if (mn == nstar) {
        float* __restrict__ erow = energies + (size_t)b * L_DIM + l0 + hi * 8;
#pragma unroll
        for (int r = 0; r < 8; ++r) {
            erow[r] = acc[r];
        }
    }
}

// ---------------------------------------------------------------------------
// Pass 2: row softmax over L=4096 per batch. One 256-thread block per b.
// ---------------------------------------------------------------------------
__global__ __launch_bounds__(256)
void softmax_kernel(const float* __restrict__ energies,
                    float* __restrict__ out) {
    const int b = blockIdx.x;
    const int t = threadIdx.x;            // 0..255
    const float* __restrict__ row = energies + (size_t)b * L_DIM;

    float vals[16];
    float lmax = -INFINITY;
#pragma unroll
    for (int i = 0; i < 16; ++i) {
        vals[i] = row[t + i * 256];
        lmax = fmaxf(lmax, vals[i]);
    }

    __shared__ float red[256];
    red[t] = lmax;
    __syncthreads();
#pragma unroll
    for (int s = 128; s > 0; s >>= 1) {
        if (t < s) red[t] = fmaxf(red[t], red[t + s]);
        __syncthreads();
    }
    const float m = red[0];
    __syncthreads();

    float lsum = 0.f;
#pragma unroll
    for (int i = 0; i < 16; ++i) {
        vals[i] = __expf(vals[i] - m);
        lsum += vals[i];
    }
    red[t] = lsum;
    __syncthreads();
#pragma unroll
    for (int s = 128; s > 0; s >>= 1) {
        if (t < s) red[t] += red[t + s];
        __syncthreads();
    }
    const float inv = 1.0f / red[0];

    float* __restrict__ orow = out + (size_t)b * L_DIM;
#pragma unroll
    for (int i = 0; i < 16; ++i) {
        orow[t + i * 256] = vals[i] * inv;
    }
}

// ---------------------------------------------------------------------------
// Launch: d_in[0] = hidden [H, B] f32, d_in[1] = encoder_outputs [L, B, H] f32
// d_out = attn [B, 1, L] f32. d_ws holds energies [B, L] f32 (512 KB).
// ---------------------------------------------------------------------------
extern "C" void kernel_launch(void* const* d_in, const int* in_sizes, int n_in,
                              void* d_out, int out_size, void* d_ws, size_t ws_size,
                              hipStream_t stream) {
    const float* hidden = (const float*)d_in[0];
    const float* enc    = (const float*)d_in[1];
    float* out          = (float*)d_out;
    float* energies     = (float*)d_ws;

    // 8192 waves total: 32 batches * 256 l-tiles; 8 waves per 256-thread block.
    energies_wmma_kernel<<<1024, 256, 0, stream>>>(hidden, enc, energies);
    softmax_kernel<<<B_DIM, 256, 0, stream>>>(energies, out);
}